// MeanSquaredError2_57629871178021
// MI455X (gfx1250) — compile-verified
//
#include <hip/hip_runtime.h>

typedef float v2f __attribute__((ext_vector_type(2)));
typedef float v4f __attribute__((ext_vector_type(4)));
typedef float v8f __attribute__((ext_vector_type(8)));

// Wave32 all-lanes add-reduce via ds_swizzle (group-of-32 XOR mode:
// offset[14:10]=xor_mask, [9:5]=or_mask=0, [4:0]=and_mask=0x1f).
__device__ __forceinline__ float wave_reduce_add(float x) {
  int xi;
  xi = __builtin_amdgcn_ds_swizzle(__float_as_int(x), 0x401f);  // xor 16
  x += __int_as_float(xi);
  xi = __builtin_amdgcn_ds_swizzle(__float_as_int(x), 0x201f);  // xor 8
  x += __int_as_float(xi);
  xi = __builtin_amdgcn_ds_swizzle(__float_as_int(x), 0x101f);  // xor 4
  x += __int_as_float(xi);
  xi = __builtin_amdgcn_ds_swizzle(__float_as_int(x), 0x081f);  // xor 2
  x += __int_as_float(xi);
  xi = __builtin_amdgcn_ds_swizzle(__float_as_int(x), 0x041f);  // xor 1
  x += __int_as_float(xi);
  return x;
}

__device__ __forceinline__ v2f square_pair(v4f x) {
  v2f a;
  a.x = x.x * x.x + x.y * x.y;
  a.y = x.z * x.z + x.w * x.w;
  return a;
}

__device__ __forceinline__ v8f wmma_rowsum(v2f a, v2f ones, v8f c) {
  // D[m,n] = sum_k A[m,k]*1 + C[m,n] : matrix-pipe add-reduce of 64 f32/wave.
  return __builtin_amdgcn_wmma_f32_16x16x4_f32(
      /*neg_a=*/false, a, /*neg_b=*/false, ones,
      /*c_mod=*/(short)0, c, /*reuse_a=*/false, /*reuse_b=*/false);
}

// ---------------------------------------------------------------------------
// Kernel 1: sum(h^2) over 45M floats.
//   Stream h as non-temporal b128 loads (one-pass, don't rinse L2), square and
//   pair per lane, reduce through V_WMMA_F32_16X16X4_F32 with B = ones into
//   four independent 16x16 f32 accumulators (no WMMA->WMMA RAW chain, 4 loads
//   in flight per wave). All WMMAs execute at full EXEC: the main/cleanup
//   loops have wave-uniform bounds with no predication; only the single tail
//   iteration predicates its load (never the WMMA).
// ---------------------------------------------------------------------------
__global__ __launch_bounds__(256) void sumsq_wmma_kernel(
    const v4f* __restrict__ h4, long long n4, float* __restrict__ ws) {
  const int lane = threadIdx.x & 31;
  const long long g = (long long)blockIdx.x * blockDim.x + threadIdx.x;
  long long base = g - lane;  // wave-uniform chunk start
  const long long stride = (long long)gridDim.x * blockDim.x;

  v8f acc0 = {}, acc1 = {}, acc2 = {}, acc3 = {};
  const v2f ones = {1.0f, 1.0f};

  // Main loop: 4 grid-strides per trip, no predication, 4 loads in flight.
  for (; base + 3 * stride + 32 <= n4; base += 4 * stride) {
    long long i0 = base + lane;
    v4f x0 = __builtin_nontemporal_load(&h4[i0]);
    v4f x1 = __builtin_nontemporal_load(&h4[i0 + stride]);
    v4f x2 = __builtin_nontemporal_load(&h4[i0 + 2 * stride]);
    v4f x3 = __builtin_nontemporal_load(&h4[i0 + 3 * stride]);
    acc0 = wmma_rowsum(square_pair(x0), ones, acc0);
    acc1 = wmma_rowsum(square_pair(x1), ones, acc1);
    acc2 = wmma_rowsum(square_pair(x2), ones, acc2);
    acc3 = wmma_rowsum(square_pair(x3), ones, acc3);
  }
  // Cleanup: single-stride full-wave iterations, still unpredicated.
  for (; base + 32 <= n4; base += stride) {
    v4f x = __builtin_nontemporal_load(&h4[base + lane]);
    acc0 = wmma_rowsum(square_pair(x), ones, acc0);
  }
  // Tail: one partial wave; only the load is predicated, WMMA at full EXEC.
  if (base < n4) {  // wave-uniform condition
    long long idx = base + lane;
    v2f a = {0.0f, 0.0f};
    if (idx < n4) a = square_pair(__builtin_nontemporal_load(&h4[idx]));
    acc0 = wmma_rowsum(a, ones, acc0);
  }

  acc0 += acc1;
  acc2 += acc3;
  acc0 += acc2;
  float s = acc0[0] + acc0[1] + acc0[2] + acc0[3] +
            acc0[4] + acc0[5] + acc0[6] + acc0[7];
  s = wave_reduce_add(s);
  // Each row-sum is replicated across 16 N-columns -> exact /16 rescale.
  if (lane == 0) atomicAdd(ws, s * 0.0625f);
}

// ---------------------------------------------------------------------------
// Kernel 2: one-hot correction  sum_visible (1 - 2*h[b,j,xi,yi]).
//   xi = clamp(trunc(t0*14),0,13), yi = clamp(trunc(t1*14),0,13).
// ---------------------------------------------------------------------------
__global__ __launch_bounds__(256) void joint_corr_kernel(
    const float* __restrict__ h, const float* __restrict__ t,
    const int* __restrict__ v, int njoints, float* __restrict__ ws) {
  int i = blockIdx.x * blockDim.x + threadIdx.x;
  float c = 0.0f;
  if (i < njoints && v[i] == 1) {
    int xi = (int)(t[2 * i + 0] * 14.0f);
    int yi = (int)(t[2 * i + 1] * 14.0f);
    xi = min(max(xi, 0), 13);
    yi = min(max(yi, 0), 13);
    c = 1.0f - 2.0f * h[(long long)i * 196 + xi * 14 + yi];
  }
  c = wave_reduce_add(c);
  if ((threadIdx.x & 31) == 0) atomicAdd(ws + 1, c);
}

// ---------------------------------------------------------------------------
// Kernel 3: finalize  out = (sumsq + corr) / (h.size/2)
// ---------------------------------------------------------------------------
__global__ void finalize_kernel(const float* __restrict__ ws,
                                float* __restrict__ out, float n_half) {
  out[0] = (ws[0] + ws[1]) / n_half;
}

extern "C" void kernel_launch(void* const* d_in, const int* in_sizes, int n_in,
                              void* d_out, int out_size, void* d_ws,
                              size_t ws_size, hipStream_t stream) {
  // inputs: d_in[0]=o (unused), d_in[1]=h, d_in[2]=t, d_in[3]=v
  const float* h = (const float*)d_in[1];
  const float* t = (const float*)d_in[2];
  const int* v = (const int*)d_in[3];
  long long n_h = (long long)in_sizes[1];  // B*NJ*14*14 (multiple of 4)
  int njoints = in_sizes[3];               // B*NJ

  float* ws = (float*)d_ws;
  hipMemsetAsync(ws, 0, 2 * sizeof(float), stream);  // graph-capture safe

  long long n4 = n_h / 4;
  // 16K waves saturate 23.3 TB/s; each lane streams ~21 b128 loads.
  sumsq_wmma_kernel<<<2048, 256, 0, stream>>>((const v4f*)h, n4, ws);
  joint_corr_kernel<<<(njoints + 255) / 256, 256, 0, stream>>>(h, t, v,
                                                               njoints, ws);
  finalize_kernel<<<1, 1, 0, stream>>>(ws, (float*)d_out, (float)(n_h / 2));
}